// UncertaintyRecommender_1958505087510
// MI455X (gfx1250) — compile-verified
//
#include <hip/hip_runtime.h>
#include <hip/hip_bf16.h>

typedef __attribute__((ext_vector_type(2))) float v2f;
typedef __attribute__((ext_vector_type(8))) float v8f;

#define N_USER   100000
#define N_MOVIE  20000
#define D_USER   32
#define D_MOVIE  64
#define HDIM     64
#define E_EDGES  2000000
#define B_LABELS 200000

// ---------------------------------------------------------------------------
// Zero a float buffer (graph-capture-safe replacement for hipMemsetAsync).
// ---------------------------------------------------------------------------
__global__ void zero_kernel(float* __restrict__ p, long n) {
  long i = (long)blockIdx.x * blockDim.x + threadIdx.x;
  long stride = (long)gridDim.x * blockDim.x;
  for (; i < n; i += stride) p[i] = 0.0f;
}

// ---------------------------------------------------------------------------
// Edge aggregation: one wave32 per edge. Coalesced gather of the source row,
// f32 atomics into the destination row (L2-resident). Optionally counts
// degree (only needed once; graph structure identical across layers).
// ---------------------------------------------------------------------------
template<int D, bool DO_CNT>
__global__ void agg_kernel(const int* __restrict__ src, const int* __restrict__ dst,
                           const float* __restrict__ x, float* __restrict__ agg,
                           float* __restrict__ cnt, int nE) {
  int gid  = blockIdx.x * blockDim.x + threadIdx.x;
  int e    = gid >> 5;
  int lane = gid & 31;
  if (e >= nE) return;
  int s = src[e];
  int d = dst[e];
  const float* xs = x + (long)s * D;
  float* ad = agg + (long)d * D;
  atomicAdd(&ad[lane], xs[lane]);
  if (D == 64) atomicAdd(&ad[lane + 32], xs[lane + 32]);
  if (DO_CNT && lane == 0) atomicAdd(&cnt[d], 1.0f);
}

// ---------------------------------------------------------------------------
// SAGE node update:  out[n] = act( (agg[n]/max(cnt,1)) @ Wl + b + x[n] @ Wr )
// 128 threads = 4 waves; block computes 16 rows x 64 cols. Each wave owns one
// 16x16 output tile and walks K in steps of 4 with V_WMMA_F32_16X16X4_F32
// (pure fp32 matrix core — keeps reference precision).
//
// Fragment layouts (ISA 7.12.2, 32-bit 16x4 A):
//   A: lane&15 = M, (lane>>4)*2 + vgpr = K   -> a = {A[m][k0+kb], A[m][k0+kb+1]}
//   B: assumed transpose-symmetric: lane&15 = N, same K mapping
//   C/D: lane&15 = N, M = vgpr + 8*(lane>>4)
// ---------------------------------------------------------------------------
template<int DL, int DR, bool RELU>
__global__ __launch_bounds__(128)
void sage_node_kernel(const float* __restrict__ agg,
                      const float* __restrict__ cnt,
                      const float* __restrict__ x,
                      const float* __restrict__ Wl,
                      const float* __restrict__ bias,
                      const float* __restrict__ Wr,
                      float* __restrict__ out, int N) {
  constexpr int F = DL + DR;
  __shared__ float sA[16 * F];

  int rowbase = blockIdx.x * 16;

  // Stage 16 node rows into LDS: [mean(DL) | x(DR)], mean pre-scaled by 1/cnt.
  for (int i = threadIdx.x; i < 16 * F; i += 128) {
    int r = i / F;
    int f = i % F;
    int node = rowbase + r;
    float v = 0.0f;
    if (node < N) {
      if (f < DL) v = agg[(long)node * DL + f] * (1.0f / fmaxf(cnt[node], 1.0f));
      else        v = x[(long)node * DR + (f - DL)];
    }
    sA[i] = v;
  }
  __syncthreads();

  int lane = threadIdx.x & 31;
  int wave = threadIdx.x >> 5;        // 0..3 -> output column tile
  int col  = wave * 16 + (lane & 15); // output column 0..63
  int row  = lane & 15;               // A-fragment row
  int kb   = (lane >> 4) * 2;         // K sub-offset per lane half

  v8f c = {};

  // mean @ Wl
#pragma unroll
  for (int k0 = 0; k0 < DL; k0 += 4) {
    v2f a, b;
    a.x = sA[row * F + k0 + kb];
    a.y = sA[row * F + k0 + kb + 1];
    b.x = Wl[(k0 + kb) * HDIM + col];
    b.y = Wl[(k0 + kb + 1) * HDIM + col];
    c = __builtin_amdgcn_wmma_f32_16x16x4_f32(false, a, false, b, (short)0, c,
                                              false, false);
  }
  // x @ Wr
#pragma unroll
  for (int k0 = 0; k0 < DR; k0 += 4) {
    v2f a, b;
    a.x = sA[row * F + DL + k0 + kb];
    a.y = sA[row * F + DL + k0 + kb + 1];
    b.x = Wr[(k0 + kb) * HDIM + col];
    b.y = Wr[(k0 + kb + 1) * HDIM + col];
    c = __builtin_amdgcn_wmma_f32_16x16x4_f32(false, a, false, b, (short)0, c,
                                              false, false);
  }

  float bv = bias[col];
  int mbase = (lane >> 4) * 8;
#pragma unroll
  for (int v = 0; v < 8; v++) {
    int node = rowbase + mbase + v;
    if (node < N) {
      float val = c[v] + bv;
      if (RELU) val = fmaxf(val, 0.0f);
      out[(long)node * HDIM + col] = val;
    }
  }
}

// ---------------------------------------------------------------------------
// Head: gather z_user[row], z_movie[col], concat(128) @ W_lin(128x2) + b,
// mu = out0, sigma = softplus(out1) + 1e-6 (numerically stable form).
// ---------------------------------------------------------------------------
__global__ void head_kernel(const float* __restrict__ z_user,
                            const float* __restrict__ z_movie,
                            const int* __restrict__ lrow,
                            const int* __restrict__ lcol,
                            const float* __restrict__ W_lin,
                            const float* __restrict__ b_lin,
                            float* __restrict__ out) {
  __shared__ float sW[256];
  for (int i = threadIdx.x; i < 256; i += blockDim.x) sW[i] = W_lin[i];
  __syncthreads();

  int i = blockIdx.x * blockDim.x + threadIdx.x;
  if (i >= B_LABELS) return;

  const float* zu = z_user + (long)lrow[i] * HDIM;
  const float* zm = z_movie + (long)lcol[i] * HDIM;
  float m = b_lin[0];
  float s = b_lin[1];
#pragma unroll 8
  for (int k = 0; k < HDIM; k++) {
    float v = zu[k];
    m = fmaf(v, sW[2 * k], m);
    s = fmaf(v, sW[2 * k + 1], s);
  }
#pragma unroll 8
  for (int k = 0; k < HDIM; k++) {
    float v = zm[k];
    m = fmaf(v, sW[2 * (HDIM + k)], m);
    s = fmaf(v, sW[2 * (HDIM + k) + 1], s);
  }
  out[i] = m;
  float sp = fmaxf(s, 0.0f) + log1pf(expf(-fabsf(s)));
  out[B_LABELS + i] = sp + 1e-6f;
}

// ---------------------------------------------------------------------------
extern "C" void kernel_launch(void* const* d_in, const int* in_sizes, int n_in,
                              void* d_out, int out_size, void* d_ws, size_t ws_size,
                              hipStream_t stream) {
  const float* x_user  = (const float*)d_in[0];
  const float* x_movie = (const float*)d_in[1];
  const int* src_um    = (const int*)d_in[2];
  const int* dst_um    = (const int*)d_in[3];
  const int* src_mu    = (const int*)d_in[4];
  const int* dst_mu    = (const int*)d_in[5];
  const int* label_row = (const int*)d_in[6];
  const int* label_col = (const int*)d_in[7];
  const float* W1_um_l = (const float*)d_in[8];
  const float* b1_um   = (const float*)d_in[9];
  const float* W1_um_r = (const float*)d_in[10];
  const float* W1_mu_l = (const float*)d_in[11];
  const float* b1_mu   = (const float*)d_in[12];
  const float* W1_mu_r = (const float*)d_in[13];
  const float* W2_um_l = (const float*)d_in[14];
  const float* b2_um   = (const float*)d_in[15];
  const float* W2_um_r = (const float*)d_in[16];
  const float* W2_mu_l = (const float*)d_in[17];
  const float* b2_mu   = (const float*)d_in[18];
  const float* W2_mu_r = (const float*)d_in[19];
  const float* W_lin   = (const float*)d_in[20];
  const float* b_lin   = (const float*)d_in[21];

  // Workspace layout (floats), ~93 MB total.
  float* ws        = (float*)d_ws;
  float* agg_user  = ws;                                   // N_USER  * 64
  float* agg_movie = agg_user + (long)N_USER * HDIM;       // N_MOVIE * 64
  float* cnt_user  = agg_movie + (long)N_MOVIE * HDIM;     // N_USER
  float* cnt_movie = cnt_user + N_USER;                    // N_MOVIE
  float* h_user    = cnt_movie + N_MOVIE;                  // N_USER  * 64
  float* h_movie   = h_user + (long)N_USER * HDIM;         // N_MOVIE * 64
  float* z_user    = h_movie + (long)N_MOVIE * HDIM;       // N_USER  * 64
  float* z_movie   = z_user + (long)N_USER * HDIM;         // N_MOVIE * 64

  const int aggBlocks = (E_EDGES * 32 + 255) / 256;  // one wave32 per edge

  // ---- Layer 1 ----
  long zeroN1 = (long)N_USER * HDIM + (long)N_MOVIE * HDIM + N_USER + N_MOVIE;
  zero_kernel<<<2048, 256, 0, stream>>>(agg_user, zeroN1);

  agg_kernel<32, true><<<aggBlocks, 256, 0, stream>>>(src_um, dst_um, x_user,
                                                      agg_movie, cnt_movie, E_EDGES);
  agg_kernel<64, true><<<aggBlocks, 256, 0, stream>>>(src_mu, dst_mu, x_movie,
                                                      agg_user, cnt_user, E_EDGES);

  sage_node_kernel<32, 64, true><<<(N_MOVIE + 15) / 16, 128, 0, stream>>>(
      agg_movie, cnt_movie, x_movie, W1_um_l, b1_um, W1_um_r, h_movie, N_MOVIE);
  sage_node_kernel<64, 32, true><<<(N_USER + 15) / 16, 128, 0, stream>>>(
      agg_user, cnt_user, x_user, W1_mu_l, b1_mu, W1_mu_r, h_user, N_USER);

  // ---- Layer 2 (same graph -> reuse counts) ----
  long zeroN2 = (long)N_USER * HDIM + (long)N_MOVIE * HDIM;
  zero_kernel<<<2048, 256, 0, stream>>>(agg_user, zeroN2);

  agg_kernel<64, false><<<aggBlocks, 256, 0, stream>>>(src_um, dst_um, h_user,
                                                       agg_movie, nullptr, E_EDGES);
  agg_kernel<64, false><<<aggBlocks, 256, 0, stream>>>(src_mu, dst_mu, h_movie,
                                                       agg_user, nullptr, E_EDGES);

  sage_node_kernel<64, 64, false><<<(N_MOVIE + 15) / 16, 128, 0, stream>>>(
      agg_movie, cnt_movie, h_movie, W2_um_l, b2_um, W2_um_r, z_movie, N_MOVIE);
  sage_node_kernel<64, 64, false><<<(N_USER + 15) / 16, 128, 0, stream>>>(
      agg_user, cnt_user, h_user, W2_mu_l, b2_mu, W2_mu_r, z_user, N_USER);

  // ---- Head ----
  head_kernel<<<(B_LABELS + 255) / 256, 256, 0, stream>>>(
      z_user, z_movie, label_row, label_col, W_lin, b_lin, (float*)d_out);
}